// GraphConv_72060961292432
// MI455X (gfx1250) — compile-verified
//
#include <hip/hip_runtime.h>

typedef __attribute__((ext_vector_type(2))) float v2f;
typedef __attribute__((ext_vector_type(8))) float v8f;

constexpr int Bb   = 2;
constexpr int CIN  = 128;
constexpr int COUT = 128;
constexpr int NN   = 100000;   // divisible by 16
constexpr int EE   = 800000;
constexpr float BN_EPS = 1e-5f;
constexpr float SLOPE  = 0.3f;

// ---------------------------------------------------------------------------
// GEMM: F_v = W_v @ X + b_v  (stored (B, C, N))
//       F_n = W_n @ X + b_n  (stored transposed (B, N, C) for coalesced edges)
// grid (NN/16, B), block 256 = 8 waves. Each wave owns a 16(M)x16(N) tile.
// X tile (128 K x 16 N) staged in LDS once, shared by all 8 waves and both
// weight matrices.
// ---------------------------------------------------------------------------
__global__ __launch_bounds__(256)
void gemm_dual_kernel(const float* __restrict__ X,    // (B, CIN, N)
                      const float* __restrict__ Wv,   // (COUT, CIN) row-major
                      const float* __restrict__ bv,   // (COUT)
                      const float* __restrict__ Wn,   // (COUT, CIN)
                      const float* __restrict__ bn,   // (COUT)
                      float* __restrict__ Fv,         // (B, COUT, N)
                      float* __restrict__ Fnt)        // (B, N, COUT)
{
    constexpr int LDSP = 17;                 // padded stride (bank-conflict free)
    __shared__ float xs[CIN * LDSP];

    const int n0  = blockIdx.x * 16;
    const int b   = blockIdx.y;
    const int tid = threadIdx.x;

    // ---- stage X tile: 128 rows (k) x 16 cols (n) ----
    const float* Xb = X + (size_t)b * CIN * NN;
    #pragma unroll
    for (int i = 0; i < 8; ++i) {
        int idx = tid + i * 256;             // 2048 elements total
        int k = idx >> 4;
        int n = idx & 15;
        xs[k * LDSP + n] = Xb[(size_t)k * NN + n0 + n];
    }
    __syncthreads();

    const int wave = tid >> 5;
    const int lane = tid & 31;
    const int hi   = lane >> 4;              // which half of the wave
    const int l16  = lane & 15;
    const int m0   = wave * 16;              // this wave's output-row tile

    // A-fragment addressing: lanes 0-15 -> K={0,1}, lanes 16-31 -> K={2,3}
    const int arow = m0 + l16;
    const int koff = hi * 2;

    const float* wvrow = Wv + arow * CIN;
    const float* wnrow = Wn + arow * CIN;

    v8f accv = {};
    v8f accn = {};

    #pragma unroll
    for (int kc = 0; kc < CIN; kc += 4) {
        // A fragments (float2, 8-byte aligned: kc%4==0, koff even)
        v2f av = *reinterpret_cast<const v2f*>(wvrow + kc + koff);
        v2f an = *reinterpret_cast<const v2f*>(wnrow + kc + koff);
        // B fragment from LDS: column = l16, rows K = kc+koff, kc+koff+1
        v2f bx;
        bx.x = xs[(kc + koff)     * LDSP + l16];
        bx.y = xs[(kc + koff + 1) * LDSP + l16];
        // D = A*B + C  (fp32 WMMA, 8 args: neg_a, A, neg_b, B, c_mod, C, reuse)
        accv = __builtin_amdgcn_wmma_f32_16x16x4_f32(false, av, false, bx,
                                                     (short)0, accv, false, false);
        accn = __builtin_amdgcn_wmma_f32_16x16x4_f32(false, an, false, bx,
                                                     (short)0, accn, false, false);
    }

    // ---- epilogue: bias add + dual-layout store ----
    // D layout: VGPR r, lane<16 -> (M=r, N=l16); lane>=16 -> (M=r+8, N=l16)
    const int ncol = n0 + l16;
    float* fvb  = Fv  + (size_t)b * COUT * NN;
    float* fntb = Fnt + (size_t)b * NN * COUT;
    #pragma unroll
    for (int r = 0; r < 8; ++r) {
        int row = m0 + r + hi * 8;
        float vv = accv[r] + bv[row];
        float vn = accn[r] + bn[row];
        fvb[(size_t)row * NN + ncol]    = vv;   // (B, C, N)
        fntb[(size_t)ncol * COUT + row] = vn;   // (B, N, C)
    }
}

// ---------------------------------------------------------------------------
// Edge degree counts (float, so the mean divide is a single fmax+div later)
// ---------------------------------------------------------------------------
__global__ __launch_bounds__(256)
void count_kernel(const int* __restrict__ ridx, float* __restrict__ counts)
{
    int e = blockIdx.x * 256 + threadIdx.x;
    if (e < EE) atomicAdd(&counts[ridx[e]], 1.0f);
}

// ---------------------------------------------------------------------------
// Scatter-add of gathered neighbor features.
// (B,N,C) layout => 128 contiguous floats per edge: fully coalesced gather
// and coalesced global_atomic_add_f32 scatter.
// grid (EE, B), block 128 (one channel per thread).
// ---------------------------------------------------------------------------
__global__ __launch_bounds__(128)
void scatter_kernel(const float* __restrict__ Fnt,   // (B, N, C)
                    const int* __restrict__ gidx,
                    const int* __restrict__ ridx,
                    float* __restrict__ Sums)        // (B, N, C)
{
    const int e = blockIdx.x;
    const int b = blockIdx.y;
    const int c = threadIdx.x;
    const int g = gidx[e];
    const int r = ridx[e];
    float v = Fnt[((size_t)b * NN + g) * COUT + c];
    atomicAdd(&Sums[((size_t)b * NN + r) * COUT + c], v);
}

// ---------------------------------------------------------------------------
// out = Sums/max(count,1) + F_v   (written to d_out pre-BN)
// plus per-channel sum / sumsq partials (block tree-reduce -> atomics).
// grid (ceil(N/256), C, B); all threads in a block share one channel.
// ---------------------------------------------------------------------------
__global__ __launch_bounds__(256)
void combine_stats_kernel(const float* __restrict__ Fv,     // (B, C, N)
                          const float* __restrict__ Sums,   // (B, N, C)
                          const float* __restrict__ counts, // (N)
                          float* __restrict__ out,          // (B, C, N)
                          float* __restrict__ chanSum,      // (C)
                          float* __restrict__ chanSq)       // (C)
{
    const int n = blockIdx.x * 256 + threadIdx.x;
    const int c = blockIdx.y;
    const int b = blockIdx.z;

    float t = 0.0f;
    const bool valid = (n < NN);
    if (valid) {
        float cnt  = counts[n];
        float mean = Sums[((size_t)b * NN + n) * COUT + c] / fmaxf(cnt, 1.0f);
        t = mean + Fv[((size_t)b * COUT + c) * NN + n];
        out[((size_t)b * COUT + c) * NN + n] = t;
    }

    __shared__ float s1[256];
    __shared__ float s2[256];
    s1[threadIdx.x] = valid ? t     : 0.0f;
    s2[threadIdx.x] = valid ? t * t : 0.0f;
    __syncthreads();
    #pragma unroll
    for (int s = 128; s > 0; s >>= 1) {
        if (threadIdx.x < s) {
            s1[threadIdx.x] += s1[threadIdx.x + s];
            s2[threadIdx.x] += s2[threadIdx.x + s];
        }
        __syncthreads();
    }
    if (threadIdx.x == 0) {
        atomicAdd(&chanSum[c], s1[0]);
        atomicAdd(&chanSq[c],  s2[0]);
    }
}

// ---------------------------------------------------------------------------
// Finalize BN stats into per-channel scale/shift. 1 block, 128 threads.
// ---------------------------------------------------------------------------
__global__ __launch_bounds__(128)
void stats_finalize_kernel(const float* __restrict__ chanSum,
                           const float* __restrict__ chanSq,
                           const float* __restrict__ gamma,
                           const float* __restrict__ beta,
                           float* __restrict__ scale,
                           float* __restrict__ shift)
{
    const int c = threadIdx.x;
    const float inv = 1.0f / (float)((size_t)Bb * NN);
    float mu  = chanSum[c] * inv;
    float var = chanSq[c] * inv - mu * mu;
    float s   = gamma[c] * rsqrtf(var + BN_EPS);
    scale[c] = s;
    shift[c] = beta[c] - mu * s;
}

// ---------------------------------------------------------------------------
// In-place normalize + LeakyReLU(0.3) over d_out.
// ---------------------------------------------------------------------------
__global__ __launch_bounds__(256)
void norm_act_kernel(float* __restrict__ out,
                     const float* __restrict__ scale,
                     const float* __restrict__ shift)
{
    const size_t total = (size_t)Bb * COUT * NN;
    size_t i = (size_t)blockIdx.x * 256 + threadIdx.x;
    if (i >= total) return;
    int c = (int)((i / NN) % COUT);
    float v = out[i] * scale[c] + shift[c];
    out[i] = (v > 0.0f) ? v : SLOPE * v;
}

// ---------------------------------------------------------------------------
// Host-side launch
// ---------------------------------------------------------------------------
extern "C" void kernel_launch(void* const* d_in, const int* in_sizes, int n_in,
                              void* d_out, int out_size, void* d_ws, size_t ws_size,
                              hipStream_t stream)
{
    (void)in_sizes; (void)n_in; (void)out_size; (void)ws_size;

    const float* X     = (const float*)d_in[0];  // in_features (B, CIN, N)
    const float* Wv    = (const float*)d_in[1];
    const float* bv    = (const float*)d_in[2];
    const float* Wn    = (const float*)d_in[3];
    const float* bn    = (const float*)d_in[4];
    const float* gamma = (const float*)d_in[5];
    const float* beta  = (const float*)d_in[6];
    const int*   ridx  = (const int*)d_in[7];    // reduce_index
    const int*   gidx  = (const int*)d_in[8];    // gather_index

    float* out = (float*)d_out;

    // Workspace layout (floats)
    const size_t szFCN = (size_t)Bb * COUT * NN; // 25.6M
    float* Fv      = (float*)d_ws;
    float* Fnt     = Fv  + szFCN;
    float* Sums    = Fnt + szFCN;                // zeroed each call
    float* counts  = Sums + szFCN;               // zeroed (N)
    float* chanSum = counts + NN;                // zeroed (C)
    float* chanSq  = chanSum + COUT;             // zeroed (C)
    float* scale   = chanSq + COUT;              // (C)
    float* shift   = scale + COUT;               // (C)

    // Zero the accumulation region [Sums .. scale) in one async memset
    size_t zeroBytes = (szFCN + NN + 2 * COUT) * sizeof(float);
    hipMemsetAsync(Sums, 0, zeroBytes, stream);

    // 1) dual GEMM with fp32 WMMA
    gemm_dual_kernel<<<dim3(NN / 16, Bb), 256, 0, stream>>>(X, Wv, bv, Wn, bn, Fv, Fnt);

    // 2) degree counts
    count_kernel<<<dim3((EE + 255) / 256), 256, 0, stream>>>(ridx, counts);

    // 3) edge gather + scatter-add (coalesced over channel dim)
    scatter_kernel<<<dim3(EE, Bb), 128, 0, stream>>>(Fnt, gidx, ridx, Sums);

    // 4) mean + F_v, write pre-BN output, accumulate channel stats
    combine_stats_kernel<<<dim3((NN + 255) / 256, COUT, Bb), 256, 0, stream>>>(
        Fv, Sums, counts, out, chanSum, chanSq);

    // 5) finalize BN scale/shift
    stats_finalize_kernel<<<1, 128, 0, stream>>>(chanSum, chanSq, gamma, beta, scale, shift);

    // 6) normalize + LeakyReLU, in place
    size_t total = (size_t)Bb * COUT * NN;
    norm_act_kernel<<<dim3((unsigned)((total + 255) / 256)), 256, 0, stream>>>(out, scale, shift);
}